// Attn_45423574123082
// MI455X (gfx1250) — compile-verified
//
#include <hip/hip_runtime.h>
#include <hip/hip_bf16.h>
#include <math.h>

typedef __attribute__((ext_vector_type(2))) float v2f;
typedef __attribute__((ext_vector_type(8))) float v8f;

#define B_SZ   128
#define S_SZ   1024
#define D_SZ   64      // SIZE1
#define N_SZ   2048
#define K_SZ   512     // SIZE0

// ---------------------------------------------------------------------------
// Stage 1: v[b,s] = sum_d W[d,s] * temp[b,d];  c[b] = sum_d bias[d]*temp[b,d]
// temp[b,d] = enc[b, current_index, d].  One block per batch, 512 threads.
// ---------------------------------------------------------------------------
__global__ __launch_bounds__(512) void fold_kernel(
    const float* __restrict__ enc, const int* __restrict__ cur_idx,
    const float* __restrict__ W, const float* __restrict__ bias,
    float* __restrict__ v, float* __restrict__ c) {
  __shared__ __align__(16) float t[D_SZ];
  const int b  = blockIdx.x;
  const int ci = *cur_idx;
  if (threadIdx.x < D_SZ)
    t[threadIdx.x] = enc[((size_t)b * S_SZ + ci) * D_SZ + threadIdx.x];
  __syncthreads();
  const int s = threadIdx.x;
  float acc  = 0.0f;
  float cacc = 0.0f;
  #pragma unroll 8
  for (int d = 0; d < D_SZ; ++d) {
    const float td = t[d];
    acc = fmaf(W[d * K_SZ + s], td, acc);       // W is (64, 512) row-major
    if (threadIdx.x == 0) cacc = fmaf(bias[d], td, cacc);
  }
  v[(size_t)b * K_SZ + s] = acc;
  if (threadIdx.x == 0) c[b] = cacc;
}

// ---------------------------------------------------------------------------
// Stage 2: e[b,n] = neighbor[b,n,:] . v[b,:] + c[b]  via V_WMMA_F32_16X16X4_F32
// One wave per 16-neighbor tile. A = 16x4 neighbor slab, B = v[k..k+3]
// broadcast across 16 columns -> every D column holds the 16 dot products.
//   A layout (32-bit 16x4): lanes 0-15 = M rows, VGPR0/1 = K0/K1;
//                           lanes 16-31 = same M rows, VGPR0/1 = K2/K3.
//   D layout: lane j (j<16) VGPR r = (M=r, N=j); lane j+16 = (M=r+8, N=j).
// ---------------------------------------------------------------------------
__global__ __launch_bounds__(256) void energy_wmma_kernel(
    const float* __restrict__ nb, const float* __restrict__ v,
    const float* __restrict__ c, float* __restrict__ e) {
  __shared__ __align__(16) float vs[K_SZ];

  const int lane    = threadIdx.x & 31;
  const int wave    = threadIdx.x >> 5;
  const int wave_id = blockIdx.x * 8 + wave;   // 8 waves / block
  const int b       = wave_id >> 7;            // 128 tiles per batch (block-uniform)
  const int tile    = wave_id & 127;

  // Stage v[b] (2 KB) into LDS once per block (b is uniform across the block).
  for (int i = threadIdx.x; i < K_SZ; i += 256)
    vs[i] = v[(size_t)b * K_SZ + i];
  __syncthreads();

  const int   row  = tile * 16 + (lane & 15);
  const int   koff = (lane >> 4) << 1;                      // 0 or 2
  const float* arow = nb + ((size_t)b * N_SZ + row) * K_SZ + koff;

  v8f acc = {0.f, 0.f, 0.f, 0.f, 0.f, 0.f, 0.f, 0.f};

  for (int k = 0; k < K_SZ; k += 16) {
    __builtin_prefetch(arow + k + 64, 0, 0);                // stream next 256B of this row
    #pragma unroll
    for (int kk = 0; kk < 16; kk += 4) {
      const v2f a  = *(const v2f*)(arow + k + kk);          // A: (row, k+kk+koff .. +1)
      const v2f bb = *(const v2f*)(&vs[k + kk + koff]);     // B: v broadcast across columns
      acc = __builtin_amdgcn_wmma_f32_16x16x4_f32(
          /*neg_a=*/false, a, /*neg_b=*/false, bb,
          /*c_mod=*/(short)0, acc, /*reuse_a=*/false, /*reuse_b=*/false);
    }
  }

  // Column N=0 carries the results: lane 0 -> M0..7, lane 16 -> M8..15.
  const float cb = c[b];
  if ((lane & 15) == 0) {
    float* erow = e + (size_t)b * N_SZ + tile * 16 + (lane >> 4) * 8;
    #pragma unroll
    for (int i = 0; i < 8; ++i) erow[i] = acc[i] + cb;
  }
}

// ---------------------------------------------------------------------------
// Stage 3: softmax over n (2048) per batch. One block (256 thr) per batch;
// each thread caches its 8 values in registers; wave32 shfl + LDS reduction.
// ---------------------------------------------------------------------------
__global__ __launch_bounds__(256) void softmax_kernel(
    const float* __restrict__ e, float* __restrict__ out) {
  __shared__ float red[8];
  const int b = blockIdx.x;
  const float* row = e + (size_t)b * N_SZ;
  const int lane = threadIdx.x & 31;
  const int wave = threadIdx.x >> 5;

  float loc[8];
  float m = -3.402823466e38f;
  #pragma unroll
  for (int j = 0; j < 8; ++j) {
    loc[j] = row[threadIdx.x + j * 256];
    m = fmaxf(m, loc[j]);
  }
  #pragma unroll
  for (int off = 16; off >= 1; off >>= 1)
    m = fmaxf(m, __shfl_xor(m, off, 32));
  if (lane == 0) red[wave] = m;
  __syncthreads();
  float bm = red[0];
  #pragma unroll
  for (int w = 1; w < 8; ++w) bm = fmaxf(bm, red[w]);

  float s = 0.f;
  #pragma unroll
  for (int j = 0; j < 8; ++j) {
    loc[j] = __expf(loc[j] - bm);
    s += loc[j];
  }
  #pragma unroll
  for (int off = 16; off >= 1; off >>= 1)
    s += __shfl_xor(s, off, 32);
  __syncthreads();
  if (lane == 0) red[wave] = s;
  __syncthreads();
  float bs = 0.f;
  #pragma unroll
  for (int w = 0; w < 8; ++w) bs += red[w];
  const float inv = 1.0f / bs;

  #pragma unroll
  for (int j = 0; j < 8; ++j)
    out[(size_t)b * N_SZ + threadIdx.x + j * 256] = loc[j] * inv;
}

// ---------------------------------------------------------------------------
// Launch. Input order: enc, current_index, neighbor, neighbor_num, W, b.
// Workspace: v (128*512 f32) | c (128 f32) | e (128*2048 f32)  ~ 1.31 MB.
// ---------------------------------------------------------------------------
extern "C" void kernel_launch(void* const* d_in, const int* in_sizes, int n_in,
                              void* d_out, int out_size, void* d_ws, size_t ws_size,
                              hipStream_t stream) {
  const float* enc  = (const float*)d_in[0];
  const int*   cur  = (const int*)  d_in[1];
  const float* nb   = (const float*)d_in[2];
  const float* W    = (const float*)d_in[4];
  const float* bias = (const float*)d_in[5];
  float* out = (float*)d_out;

  float* v = (float*)d_ws;           // 65536 floats
  float* c = v + B_SZ * K_SZ;        // 128 floats
  float* e = c + B_SZ;               // 262144 floats

  fold_kernel<<<B_SZ, 512, 0, stream>>>(enc, cur, W, bias, v, c);
  // 128 batches * 128 tiles = 16384 waves / 8 waves per block = 2048 blocks
  energy_wmma_kernel<<<2048, 256, 0, stream>>>(nb, v, c, e);
  softmax_kernel<<<B_SZ, 256, 0, stream>>>(e, out);
}